// ArcFace_2430951489683
// MI455X (gfx1250) — compile-verified
//
#include <hip/hip_runtime.h>
#include <math.h>

// ArcFace: out = logits * S, except out[r, label[r]] = cos(arccos(clip(t)) + m) * S
// for valid labels. Memory-bound streaming scale (1.64 GB @ 23.3 TB/s ~ 70us floor).

#define S_SCALE 64.0f
#define COS_M   0.87758256189037271612f  // cos(0.5)
#define SIN_M   0.47942553860420300027f  // sin(0.5)

typedef float v4f __attribute__((ext_vector_type(4)));
typedef int   v4i __attribute__((ext_vector_type(4)));

#if __has_builtin(__builtin_amdgcn_global_load_async_to_lds_b128) && \
    __has_builtin(__builtin_amdgcn_s_wait_asynccnt)
#define USE_ASYNC 1
#else
#define USE_ASYNC 0
#endif

// Exact parameter types per hipcc diagnostic:
//   arg0: int __vector(4) addrspace(1)*   (global source, printed as "__device__")
//   arg1: int __vector(4) addrspace(3)*   (LDS destination)
typedef __attribute__((address_space(1))) v4i* gv4i_t;
typedef __attribute__((address_space(3))) v4i* lv4i_t;

// ---------------------------------------------------------------------------
// Bulk streaming scale kernel: out[i] = in[i] * 64.
// Block tile = 512 float4s (8KB). Each thread covers tile elements tid and
// tid+256, so each async issue is a dense, fully-coalesced 4KB per block.
// Double-buffered via ASYNCcnt: up to 4 outstanding async-DMA ops per wave.
// ---------------------------------------------------------------------------
__global__ __launch_bounds__(256) void arcface_scale(
    const float* __restrict__ in, float* __restrict__ out,
    int n4, long long total) {
  const int tid = threadIdx.x;

  // Scalar tail (total % 4 elements) — first few threads of the grid only.
  {
    const long long base = 4ll * (long long)n4;
    const long long t = (long long)blockIdx.x * 256 + tid;
    if (base + t < total) {
      out[base + t] = in[base + t] * S_SCALE;
    }
  }

#if USE_ASYNC
  __shared__ v4f lds[2][2][256];  // [buffer][pair][thread] = 16KB

  const int numTiles = (n4 + 511) >> 9;  // 512 float4s per tile
  const int tstride  = gridDim.x;

  auto issue = [&](int tileIdx, int b) {
    const long long e0 = (long long)tileIdx * 512 + tid;
    if (e0 < n4) {
      __builtin_amdgcn_global_load_async_to_lds_b128(
          (gv4i_t)(in + 4ll * e0), (lv4i_t)&lds[b][0][tid], 0, 0);
    }
    const long long e1 = e0 + 256;
    if (e1 < n4) {
      __builtin_amdgcn_global_load_async_to_lds_b128(
          (gv4i_t)(in + 4ll * e1), (lv4i_t)&lds[b][1][tid], 0, 0);
    }
  };

  int tile = blockIdx.x;
  int buf = 0;
  if (tile < numTiles) issue(tile, 0);

  for (; tile < numTiles; tile += tstride) {
    const int nextTile = tile + tstride;
    if (nextTile < numTiles) {
      issue(nextTile, buf ^ 1);
      __builtin_amdgcn_s_wait_asynccnt(2);  // current pair retired (in-order)
    } else {
      __builtin_amdgcn_s_wait_asynccnt(0);
    }
    const long long e0 = (long long)tile * 512 + tid;
    const long long e1 = e0 + 256;
    if (e0 < n4) {
      v4f v = lds[buf][0][tid];
      v *= S_SCALE;
      __builtin_nontemporal_store(v, (v4f*)(out + 4ll * e0));  // NT: no L2 rinse
    }
    if (e1 < n4) {
      v4f v = lds[buf][1][tid];
      v *= S_SCALE;
      __builtin_nontemporal_store(v, (v4f*)(out + 4ll * e1));
    }
    buf ^= 1;
  }
#else
  // Direct path: NT 128-bit loads/stores + explicit prefetch ahead.
  const int gid0 = blockIdx.x * 256 + tid;
  const int stride = gridDim.x * 256;
  for (int i = gid0; i < n4; i += stride) {
    const int pf = i + stride;
    if (pf < n4) __builtin_prefetch(in + 4ll * pf, 0, 0);  // global_prefetch_b8
    v4f v = __builtin_nontemporal_load((const v4f*)(in + 4ll * i));
    v *= S_SCALE;
    __builtin_nontemporal_store(v, (v4f*)(out + 4ll * i));
  }
#endif
}

// ---------------------------------------------------------------------------
// Fix-up: one thread per row; overwrite the target logit with the margin form.
// cos(arccos(x)+m) == x*cos(m) - sqrt(1-x^2)*sin(m)   (exact; arccos in [0,pi])
// Invalid rows (label == -1): reference scatters target back unchanged, which
// equals what the scale kernel already wrote, so we skip them.
// ---------------------------------------------------------------------------
__global__ __launch_bounds__(256) void arcface_fixup(
    const float* __restrict__ in, const int* __restrict__ labels,
    float* __restrict__ out, int B, long long C) {
  const int r = blockIdx.x * blockDim.x + threadIdx.x;
  if (r >= B) return;
  const int lbl = labels[r];
  if (lbl < 0) return;
  const long long ofs = (long long)r * C + (long long)lbl;
  const float t  = in[ofs];
  const float tc = fminf(1.0f, fmaxf(-1.0f, t));
  const float wm = tc * COS_M - sqrtf(fmaxf(0.0f, 1.0f - tc * tc)) * SIN_M;
  out[ofs] = wm * S_SCALE;
}

extern "C" void kernel_launch(void* const* d_in, const int* in_sizes, int n_in,
                              void* d_out, int out_size, void* d_ws, size_t ws_size,
                              hipStream_t stream) {
  const float* logits = (const float*)d_in[0];
  const int*   labels = (const int*)d_in[1];
  float* out = (float*)d_out;

  const long long total = (long long)in_sizes[0];  // B*C = 204,800,000
  const int Bn = in_sizes[1];                      // 4096
  const long long Cn = total / (long long)Bn;      // 50000
  const int n4 = (int)(total >> 2);                // 51,200,000 float4s

  const int blocks = 4096;  // 8 waves/block -> deep occupancy, grid-stride tiles
  arcface_scale<<<blocks, 256, 0, stream>>>(logits, out, n4, total);
  arcface_fixup<<<(Bn + 255) / 256, 256, 0, stream>>>(logits, labels, out, Bn, Cn);
}